// Translator_7060926235319
// MI455X (gfx1250) — compile-verified
//
#include <hip/hip_runtime.h>

// Problem constants (from reference)
#define NSEG      64
#define HID       1024
#define RANK      16
#define NTOK      8192
#define MAX_TILES 576   // sum over segments of ceil(count/16) <= 512 + 63

typedef __attribute__((ext_vector_type(2))) float v2f;
typedef __attribute__((ext_vector_type(8))) float v8f;

// ---------------------------------------------------------------------------
// Workspace layout (int units):
//   [0,64)        counts per segment
//   [64,129)      seg_start (exclusive prefix, 65 entries)
//   [129,193)     cursor (running scatter offsets)
//   [256,832)     tile_seg  (segment id per tile)
//   [832,1408)    tile_base (global offset in perm of tile's first token)
//   [1408]        num_tiles
//   [1472,1472+N) perm (token ids grouped by segment)
// ---------------------------------------------------------------------------

__global__ void seg_zero_kernel(int* __restrict__ counts) {
    if (threadIdx.x < NSEG) counts[threadIdx.x] = 0;
}

__global__ void seg_count_kernel(const int* __restrict__ seg,
                                 int* __restrict__ counts) {
    int n = blockIdx.x * blockDim.x + threadIdx.x;
    if (n < NTOK) atomicAdd(&counts[seg[n] & (NSEG - 1)], 1);
}

__global__ void seg_scan_kernel(const int* __restrict__ counts,
                                int* __restrict__ seg_start,
                                int* __restrict__ cursor,
                                int* __restrict__ tile_seg,
                                int* __restrict__ tile_base,
                                int* __restrict__ num_tiles) {
    // 64 segments: trivial serial scan in one thread.
    if (threadIdx.x != 0 || blockIdx.x != 0) return;
    int acc = 0, nt = 0;
    for (int s = 0; s < NSEG; ++s) {
        seg_start[s] = acc;
        cursor[s]    = acc;
        int c = counts[s];
        for (int b = 0; b < c; b += 16) {
            tile_seg[nt]  = s;
            tile_base[nt] = acc + b;
            ++nt;
        }
        acc += c;
    }
    seg_start[NSEG] = acc;
    num_tiles[0] = nt;
}

__global__ void seg_scatter_kernel(const int* __restrict__ seg,
                                   int* __restrict__ cursor,
                                   int* __restrict__ perm) {
    int n = blockIdx.x * blockDim.x + threadIdx.x;
    if (n < NTOK) {
        int pos = atomicAdd(&cursor[seg[n] & (NSEG - 1)], 1);
        perm[pos] = n;
    }
}

// ---------------------------------------------------------------------------
// Async copy of one X chunk [16 tokens x 64 hidden] (4 KB) into LDS.
// 8 wave-wide GLOBAL_LOAD_ASYNC_TO_LDS_B128: instruction i copies rows
// 2i (lanes 0-15) and 2i+1 (lanes 16-31); each half-wave moves one
// contiguous 256 B row segment. Tracked by ASYNCcnt, no VGPR round-trip.
// ---------------------------------------------------------------------------
__device__ __forceinline__ void stage_x_async(const float* __restrict__ x,
                                              const int (&myTok)[8],
                                              int cb, unsigned bufBase, int L) {
    const int colOff = (L & 15) << 2;      // float index within row
    #pragma unroll
    for (int i = 0; i < 8; ++i) {
        const float* g = x + (size_t)myTok[i] * HID + cb + colOff;
        unsigned d = bufBase + (unsigned)(512 * i) + (unsigned)(L << 4); // bytes
        asm volatile("global_load_async_to_lds_b128 %0, %1, off"
                     :: "v"(d), "v"(g)
                     : "memory");
    }
}

// ---------------------------------------------------------------------------
// Main kernel: one wave (32 lanes) per 16-token tile of one segment.
//
// WMMA f32 16x16x4 operand layouts (lane L, half h=L>>4, j=L&15):
//   A[m][k]:  lane = m + 16*(k>>1), vgpr = k&1   -> lane L supplies A[j][2h+v]
//   B[k][n]:  lane = n + 16*(k>>1), vgpr = k&1   -> lane L supplies B[2h+v][j]
//   C/D:      element (m = v + 8h, n = j) in vgpr v
// ---------------------------------------------------------------------------
__global__ __launch_bounds__(32)
void lora_wmma_kernel(const float* __restrict__ x,     // [N,H]
                      const float* __restrict__ Amat,  // [S,H,R]
                      const float* __restrict__ Bmat,  // [S,R,H]
                      const float* __restrict__ bias,  // [S,H]
                      const int*  __restrict__ tile_seg,
                      const int*  __restrict__ tile_base,
                      const int*  __restrict__ num_tiles,
                      const int*  __restrict__ seg_start,
                      const int*  __restrict__ perm,
                      float* __restrict__ out)         // [N,H]
{
    __shared__ __align__(16) float xtile[2][16 * 64];  // double-buffered X chunk
    __shared__ __align__(16) float ytile[16 * 16];     // Y: 16 tokens x 16 ranks
    __shared__ int tok[16];

    const int tile = blockIdx.x;
    if (tile >= num_tiles[0]) return;            // uniform per wave

    const int s      = tile_seg[tile];
    const int base   = tile_base[tile];
    const int segEnd = seg_start[s + 1];

    const int L = threadIdx.x;
    const int h = L >> 4;     // half-wave select (K offset pair)
    const int j = L & 15;     // M (A) / N (B,C,D) index within half

    if (L < 16) {
        int idx = base + L;
        tok[L] = (idx < segEnd) ? perm[idx] : -1;   // -1 = padding row
    }
    __syncthreads();

    // Per-lane token rows served by this lane's half in async staging:
    // instruction i covers row t = 2i + h.
    int myTok[8];
    #pragma unroll
    for (int i = 0; i < 8; ++i) {
        int tt = tok[2 * i + h];
        myTok[i] = (tt < 0) ? 0 : tt;               // clamp padding to row 0
    }

    const unsigned xBase0 = (unsigned)(uintptr_t)(void*)&xtile[0][0]; // LDS byte off
    const unsigned xBase1 = (unsigned)(uintptr_t)(void*)&xtile[1][0];

    const float* Bseg = Bmat + (size_t)s * RANK * HID;  // [16][1024]
    const float* Aseg = Amat + (size_t)s * HID * RANK;  // [1024][16]

    // ---- GEMM1: Y[16 tok x 16 rank] = X(16x1024) @ Bseg^T(1024x16) -------
    // Double-buffered: async-copy chunk c+1 while WMMAs consume chunk c.
    stage_x_async(x, myTok, 0, xBase0, L);
    v8f yacc = {0.f, 0.f, 0.f, 0.f, 0.f, 0.f, 0.f, 0.f};
    for (int cb = 0; cb < HID; cb += 64) {
        const int cur = (cb >> 6) & 1;
        const float* xcur = &xtile[cur][0];
        if (cb + 64 < HID) {
            stage_x_async(x, myTok, cb + 64, cur ? xBase0 : xBase1, L);
            asm volatile("s_wait_asynccnt 0x8" ::: "memory"); // cur chunk landed
        } else {
            asm volatile("s_wait_asynccnt 0x0" ::: "memory"); // last chunk landed
        }
        #pragma unroll
        for (int kk = 0; kk < 16; ++kk) {         // K = 64 per chunk, 4 per WMMA
            v2f a = *(const v2f*)&xcur[j * 64 + 4 * kk + 2 * h];
            v2f b = *(const v2f*)&Bseg[(size_t)j * HID + cb + 4 * kk + 2 * h];
            yacc = __builtin_amdgcn_wmma_f32_16x16x4_f32(
                false, a, false, b, (short)0, yacc, false, false);
        }
    }

    // Spill Y through LDS: C-layout -> row-major [m][r] for A-operand reuse.
    __syncthreads();
    #pragma unroll
    for (int v = 0; v < 8; ++v) {
        int m = v + 8 * h;
        ytile[m * 16 + j] = yacc[v];
    }
    __syncthreads();

    // ---- GEMM2: out[16 tok x 1024] = Y(16x16) @ Aseg^T(16x1024) + bias ---
    for (int t = 0; t < 64; ++t) {
        const int n0 = t * 16;
        float bv = bias[(size_t)s * HID + n0 + j];   // depends only on column
        v8f c = {bv, bv, bv, bv, bv, bv, bv, bv};
        #pragma unroll
        for (int kk = 0; kk < 4; ++kk) {             // K = 16 ranks, 4 per WMMA
            v2f a = *(const v2f*)&ytile[j * 16 + 4 * kk + 2 * h];
            v2f b = *(const v2f*)&Aseg[(size_t)(n0 + j) * RANK + 4 * kk + 2 * h];
            c = __builtin_amdgcn_wmma_f32_16x16x4_f32(
                false, a, false, b, (short)0, c, false, false);
        }
        #pragma unroll
        for (int v = 0; v < 8; ++v) {
            int m  = v + 8 * h;
            int tm = tok[m];
            if (tm >= 0) out[(size_t)tm * HID + n0 + j] = c[v];
        }
    }
}

// ---------------------------------------------------------------------------
extern "C" void kernel_launch(void* const* d_in, const int* in_sizes, int n_in,
                              void* d_out, int out_size, void* d_ws, size_t ws_size,
                              hipStream_t stream) {
    const float* x    = (const float*)d_in[0];   // math_hidden_states [N,H]
    const int*   seg  = (const int*)d_in[1];     // segment_ids [N] (int32 on device)
    const float* Amat = (const float*)d_in[2];   // A_matrices [S,H,R]
    const float* Bmat = (const float*)d_in[3];   // B_matrices [S,R,H]
    const float* bias = (const float*)d_in[4];   // bias [S,H]
    float* out = (float*)d_out;

    int* ws        = (int*)d_ws;
    int* counts    = ws + 0;
    int* seg_begin = ws + 64;     // 65 entries
    int* cursor    = ws + 129;
    int* tile_seg  = ws + 256;
    int* tile_base = ws + 832;
    int* num_tiles = ws + 1408;
    int* perm      = ws + 1472;

    seg_zero_kernel<<<1, 64, 0, stream>>>(counts);
    seg_count_kernel<<<NTOK / 256, 256, 0, stream>>>(seg, counts);
    seg_scan_kernel<<<1, 1, 0, stream>>>(counts, seg_begin, cursor,
                                         tile_seg, tile_base, num_tiles);
    seg_scatter_kernel<<<NTOK / 256, 256, 0, stream>>>(seg, cursor, perm);
    lora_wmma_kernel<<<MAX_TILES, 32, 0, stream>>>(
        x, Amat, Bmat, bias, tile_seg, tile_base, num_tiles, seg_begin, perm, out);
}